// SpectralConv2dUNO_23708219474528
// MI455X (gfx1250) — compile-verified
//
#include <hip/hip_runtime.h>
#include <math.h>

// SpectralConv2dUNO as 5 real GEMM stages on CDNA5 WMMA f32 16x16x4.
// B=16, CI=CO=64, H=W=256, M1=M2=32, modes km in [0,64): kx = km<32 ? km : km-64 (mod 256).

typedef float v2f __attribute__((ext_vector_type(2)));
typedef float v8f __attribute__((ext_vector_type(8)));

__device__ __forceinline__ v8f wmma4(v2f a, v2f b, v8f c) {
  // D = A(16x4) * B(4x16) + C(16x16), fp32
  return __builtin_amdgcn_wmma_f32_16x16x4_f32(false, a, false, b, (short)0, c, false, false);
}

// ---- workspace layout (floats) ----
#define OFF_T1 0u          // [256w][64n]      forward-W twiddle (thin, 1/65536 folded)
#define OFF_T2 16384u      // [512k][128n]     forward-H twiddle (2x2-block dup form)
#define OFF_T4 81920u      // [256h][128k]     inverse-H twiddle (thin A)
#define OFF_T5 114688u     // [64k][256w]      inverse-W twiddle (real output, sym factor)
#define OFF_B1 131072u     // [b][i][256h][64] stage1 out / reused as [b][o][256h][64] stage4 out
#define OFF_B2 16908288u   // [b][i][32ky][128] stage2 out
#define OFF_B3 21102592u   // [b][o][128][64]  mix out, duplicated 2x2-block form
// total floats = 29491200 (~113 MB)

__global__ void __launch_bounds__(256) fno_init_tables(float* __restrict__ ws) {
  const int t = blockIdx.x * blockDim.x + threadIdx.x;
  const float TWO_PI_256 = 6.283185307179586476925f / 256.0f;
  if (t < 16384) {                       // T1[w][2ky+c]: e^{-i 2pi ky w/256} / 65536
    int w = t >> 6, n = t & 63, ky = n >> 1, c = n & 1;
    float ang = TWO_PI_256 * (float)((ky * w) & 255);
    float v = c ? -sinf(ang) : cosf(ang);
    ws[OFF_T1 + t] = v * (1.0f / 65536.0f);
  } else if (t < 81920) {                // T2[2h+ch][2km+cn]: R(e^{-i 2pi f(km) h/256})
    int t2 = t - 16384;
    int k = t2 >> 7, n = t2 & 127;
    int h = k >> 1, ch = k & 1, km = n >> 1, cn = n & 1;
    int f = (km < 32) ? km : (km - 64);
    float ang = TWO_PI_256 * (float)((f * h) & 255);
    float cs = cosf(ang), sn = sinf(ang);
    // block [[cos,-sin],[sin,cos]]
    ws[OFF_T2 + t2] = ch ? (cn ? cs : sn) : (cn ? -sn : cs);
  } else if (t < 114688) {               // T4[h][2km+c]: (cos,sin) of +2pi f(km) h/256
    int t4 = t - 81920;
    int h = t4 >> 7, k = t4 & 127;
    int km = k >> 1, c = k & 1;
    int f = (km < 32) ? km : (km - 64);
    float ang = TWO_PI_256 * (float)((f * h) & 255);
    ws[OFF_T4 + t4] = c ? sinf(ang) : cosf(ang);
  } else if (t < 131072) {               // T5[2ky+c][w]: a_ky*(cos,-sin) of +2pi ky w/256
    int t5 = t - 114688;
    int k = t5 >> 8, w = t5 & 255;
    int ky = k >> 1, c = k & 1;
    float a = (ky == 0) ? 1.0f : 2.0f;
    float ang = TWO_PI_256 * (float)((ky * w) & 255);
    ws[OFF_T5 + t5] = c ? (-a * sinf(ang)) : (a * cosf(ang));
  }
}

// ---- Stage 1: C1[bi][h][2ky+c] = sum_w x[bi][h][w] * T1[w][n].  M=256h, K=256, N=64 ----
__global__ void __launch_bounds__(128)
fno_stage1(const float* __restrict__ x, const float* __restrict__ T1, float* __restrict__ C1) {
  const int bi   = blockIdx.x;            // b*64+i
  const int m0   = blockIdx.y * 16;       // h tile
  const int n0   = (threadIdx.x >> 5) * 16;
  const int lane = threadIdx.x & 31, half = lane >> 4, l = lane & 15;
  const int kOff = half * 2;
  const float* xrow = x  + ((size_t)bi * 256 + (m0 + l)) * 256;
  const float* bcol = T1 + (n0 + l);
  v8f acc = {};
#pragma unroll 4
  for (int k0 = 0; k0 < 256; k0 += 4) {
    const int kA = k0 + kOff;
    __builtin_prefetch(xrow + kA + 64, 0, 1);
    v2f a = *(const v2f*)(xrow + kA);
    v2f b; b.x = bcol[(size_t)kA * 64]; b.y = bcol[(size_t)(kA + 1) * 64];
    acc = wmma4(a, b, acc);
  }
  float* dst = C1 + (size_t)bi * 256 * 64;
#pragma unroll
  for (int v = 0; v < 8; ++v) {
    int m = m0 + v + half * 8;
    dst[(size_t)m * 64 + n0 + l] = acc[v];
  }
}

// ---- Stage 2: X2[bi][ky][2km+c] = sum_{2h+c} C1^T[ky][2h+c] * T2[k][n].  M=32, K=512, N=128 ----
__global__ void __launch_bounds__(256)
fno_stage2(const float* __restrict__ C1, const float* __restrict__ T2, float* __restrict__ X2) {
  const int bi   = blockIdx.x;
  const int m0   = blockIdx.y * 16;       // ky tile (0 or 16)
  const int n0   = (threadIdx.x >> 5) * 16;
  const int lane = threadIdx.x & 31, half = lane >> 4, l = lane & 15;
  const int kOff = half * 2;
  const int ky   = m0 + l;
  const float* abase = C1 + (size_t)bi * 256 * 64 + 2 * ky;  // + h*64 (+c contiguous)
  const float* bcol  = T2 + (n0 + l);
  v8f acc = {};
#pragma unroll 4
  for (int k0 = 0; k0 < 512; k0 += 4) {
    const int kA = k0 + kOff;             // even -> (h = kA/2, c=0..1 contiguous)
    v2f a = *(const v2f*)(abase + (size_t)(kA >> 1) * 64);
    v2f b; b.x = bcol[(size_t)kA * 128]; b.y = bcol[(size_t)(kA + 1) * 128];
    acc = wmma4(a, b, acc);
  }
  float* dst = X2 + (size_t)bi * 32 * 128;
#pragma unroll
  for (int v = 0; v < 8; ++v) {
    int m = m0 + v + half * 8;
    dst[(size_t)m * 128 + n0 + l] = acc[v];
  }
}

// ---- Stage 3 (mix): per mode (km,ky): Y[b][2o+c] = sum_{2i+c} X2 * R(W).  M=16, K=128, N=128
//      output written in duplicated 2x2-block form Ydup[b][o][2km+ck][2ky+cn] ----
__global__ void __launch_bounds__(256)
fno_mix(const float* __restrict__ X2,
        const float* __restrict__ wpr, const float* __restrict__ wpi,
        const float* __restrict__ wnr, const float* __restrict__ wni,
        float* __restrict__ Ydup) {
  const int ky = blockIdx.x & 31;
  const int km = blockIdx.x >> 5;         // 0..63
  const int n0 = (threadIdx.x >> 5) * 16;
  const int lane = threadIdx.x & 31, half = lane >> 4, l = lane & 15;
  const int kOff = half * 2;
  // A row: batch b = l; element k=2i+c at X2[((b*64+i)*32+ky)*128 + 2*km + c]
  const float* arow = X2 + (size_t)l * 262144 + (size_t)ky * 128 + 2 * km;
  const int n = n0 + l, o = n >> 1, cn = n & 1;
  const int m1i = (km < 32) ? km : (km - 32);
  const float* Wr = (km < 32) ? wpr : wnr;
  const float* Wi = (km < 32) ? wpi : wni;
  const size_t wbase = ((size_t)o * 32 + m1i) * 32 + ky;   // + i*65536
  v8f acc = {};
#pragma unroll 4
  for (int k0 = 0; k0 < 128; k0 += 4) {
    const int kA = k0 + kOff;             // even -> i = kA/2, c=0..1 contiguous
    const int i = kA >> 1;
    v2f a = *(const v2f*)(arow + (size_t)i * 4096);
    const size_t wi_ = wbase + (size_t)i * 65536;
    float wr = Wr[wi_], wim = Wi[wi_];
    v2f b;                                // rows 2i (Wr|Wi) and 2i+1 (-Wi|Wr)
    b.x = cn ? wim : wr;
    b.y = cn ? wr : -wim;
    acc = wmma4(a, b, acc);
  }
#pragma unroll
  for (int v = 0; v < 8; ++v) {
    const int b_ = v + half * 8;          // batch
    const float y = acc[v];
    const size_t idx0 = (((size_t)b_ * 64 + o) * 128 + 2 * km) * 64 + 2 * ky;
    if (cn == 0) { Ydup[idx0]     = y;  Ydup[idx0 + 65] = y;  }  // Yr -> (0,0),(1,1)
    else         { Ydup[idx0 + 1] = y;  Ydup[idx0 + 64] = -y; }  // Yi -> (0,1); -Yi -> (1,0)
  }
}

// ---- Stage 4: Y1[bo][h][2ky+c] = sum_k T4[h][k] * Ydup[bo][k][n].  M=256h, K=128, N=64 ----
__global__ void __launch_bounds__(128)
fno_stage4(const float* __restrict__ Ydup, const float* __restrict__ T4, float* __restrict__ Y1) {
  const int bo   = blockIdx.x;            // b*64+o
  const int m0   = blockIdx.y * 16;
  const int n0   = (threadIdx.x >> 5) * 16;
  const int lane = threadIdx.x & 31, half = lane >> 4, l = lane & 15;
  const int kOff = half * 2;
  const float* arow = T4 + (size_t)(m0 + l) * 128;
  const float* bcol = Ydup + (size_t)bo * 128 * 64 + (n0 + l);
  v8f acc = {};
#pragma unroll 4
  for (int k0 = 0; k0 < 128; k0 += 4) {
    const int kA = k0 + kOff;
    v2f a = *(const v2f*)(arow + kA);
    v2f b; b.x = bcol[(size_t)kA * 64]; b.y = bcol[(size_t)(kA + 1) * 64];
    acc = wmma4(a, b, acc);
  }
  float* dst = Y1 + (size_t)bo * 256 * 64;
#pragma unroll
  for (int v = 0; v < 8; ++v) {
    int m = m0 + v + half * 8;
    dst[(size_t)m * 64 + n0 + l] = acc[v];
  }
}

// ---- Stage 5: out[bo][h][w] = sum_k Y1[bo][h][k] * T5[k][w].  M=256h, K=64, N=256w ----
__global__ void __launch_bounds__(128)
fno_stage5(const float* __restrict__ Y1, const float* __restrict__ T5, float* __restrict__ out) {
  const int bo   = blockIdx.x;
  const int m0   = blockIdx.y * 16;
  const int n0   = (blockIdx.z * 4 + (threadIdx.x >> 5)) * 16;
  const int lane = threadIdx.x & 31, half = lane >> 4, l = lane & 15;
  const int kOff = half * 2;
  const float* arow = Y1 + ((size_t)bo * 256 + m0 + l) * 64;
  const float* bcol = T5 + (n0 + l);
  v8f acc = {};
#pragma unroll
  for (int k0 = 0; k0 < 64; k0 += 4) {
    const int kA = k0 + kOff;
    v2f a = *(const v2f*)(arow + kA);
    v2f b; b.x = bcol[(size_t)kA * 256]; b.y = bcol[(size_t)(kA + 1) * 256];
    acc = wmma4(a, b, acc);
  }
  float* dst = out + (size_t)bo * 256 * 256;
#pragma unroll
  for (int v = 0; v < 8; ++v) {
    int m = m0 + v + half * 8;
    dst[(size_t)m * 256 + n0 + l] = acc[v];
  }
}

extern "C" void kernel_launch(void* const* d_in, const int* in_sizes, int n_in,
                              void* d_out, int out_size, void* d_ws, size_t ws_size,
                              hipStream_t stream) {
  (void)in_sizes; (void)n_in; (void)out_size; (void)ws_size;
  const float* x   = (const float*)d_in[0];   // [16,64,256,256]
  const float* wpr = (const float*)d_in[1];   // [64,64,32,32]
  const float* wpi = (const float*)d_in[2];
  const float* wnr = (const float*)d_in[3];
  const float* wni = (const float*)d_in[4];
  float* ws  = (float*)d_ws;
  float* out = (float*)d_out;

  fno_init_tables<<<dim3(512), 256, 0, stream>>>(ws);
  fno_stage1<<<dim3(1024, 16),    128, 0, stream>>>(x, ws + OFF_T1, ws + OFF_B1);
  fno_stage2<<<dim3(1024, 2),     256, 0, stream>>>(ws + OFF_B1, ws + OFF_T2, ws + OFF_B2);
  fno_mix   <<<dim3(2048),        256, 0, stream>>>(ws + OFF_B2, wpr, wpi, wnr, wni, ws + OFF_B3);
  fno_stage4<<<dim3(1024, 16),    128, 0, stream>>>(ws + OFF_B3, ws + OFF_T4, ws + OFF_B1); // reuse B1
  fno_stage5<<<dim3(1024, 16, 4), 128, 0, stream>>>(ws + OFF_B1, ws + OFF_T5, out);
}